// CobraBlock_35794257444903
// MI455X (gfx1250) — compile-verified
//
#include <hip/hip_runtime.h>
#include <hip/hip_bf16.h>
#include <math.h>

typedef __attribute__((ext_vector_type(2))) float v2f;
typedef __attribute__((ext_vector_type(8))) float v8f;

__device__ __forceinline__ float softplus_f(float z) {
    return (z > 0.0f) ? (z + log1pf(expf(-z))) : log1pf(expf(z));
}

// ---------------------------------------------------------------------------
// fp32 WMMA GEMM, register-blocked 2x2:  C[M,N] = A[M,K] * W[N,K]^T (+bias)(+act)
// One wave computes a 32x32 output tile as four 16x16 WMMA tiles; each K-step
// loads 2 A-fragments + 2 B-fragments (4x global_load_b64) feeding 4 WMMAs
// (V_WMMA_F32_16X16X4_F32) -> 2x the FLOP/byte of the 1-tile version and half
// the L2 traffic on the dominant 512x1024x1024 projections.
// A row-major (lda), W row-major N x K (ldw), C row-major (ldc).
// tiles_n = N/32.  act: 0 = none, 1 = softplus.
// ---------------------------------------------------------------------------
__global__ __launch_bounds__(256)
void wmma_gemm_f32(const float* __restrict__ A, int lda,
                   const float* __restrict__ W, int ldw,
                   const float* __restrict__ bias,
                   float* __restrict__ C, int ldc,
                   int K, int tiles_n, int act) {
    const int wave = threadIdx.x >> 5;
    const int lane = threadIdx.x & 31;
    const int tile = blockIdx.x * 8 + wave;
    const int tm = tile / tiles_n;
    const int tn = tile % tiles_n;
    const int m0 = tm << 5;
    const int n0 = tn << 5;

    const int lo = lane & 15;       // M index for A frags, N index for B frags
    const int hi = lane >> 4;       // selects K pair
    const int kb = hi << 1;

    const float* __restrict__ a0row = A + (size_t)(m0 + lo) * lda + kb;
    const float* __restrict__ a1row = a0row + (size_t)16 * lda;
    const float* __restrict__ b0row = W + (size_t)(n0 + lo) * ldw + kb;
    const float* __restrict__ b1row = b0row + (size_t)16 * ldw;

    v8f acc00 = {}, acc01 = {}, acc10 = {}, acc11 = {};
#pragma unroll 4
    for (int k = 0; k < K; k += 4) {
        v2f a0 = *(const v2f*)(a0row + k);
        v2f a1 = *(const v2f*)(a1row + k);
        v2f b0 = *(const v2f*)(b0row + k);
        v2f b1 = *(const v2f*)(b1row + k);
        acc00 = __builtin_amdgcn_wmma_f32_16x16x4_f32(false, a0, false, b0, (short)0, acc00, false, false);
        acc01 = __builtin_amdgcn_wmma_f32_16x16x4_f32(false, a0, false, b1, (short)0, acc01, false, false);
        acc10 = __builtin_amdgcn_wmma_f32_16x16x4_f32(false, a1, false, b0, (short)0, acc10, false, false);
        acc11 = __builtin_amdgcn_wmma_f32_16x16x4_f32(false, a1, false, b1, (short)0, acc11, false, false);
    }

    const int c0 = n0 + lo;
    const int c1 = c0 + 16;
    const float bv0 = bias ? bias[c0] : 0.0f;
    const float bv1 = bias ? bias[c1] : 0.0f;
#pragma unroll
    for (int v = 0; v < 8; ++v) {
        const int r0 = m0 + v + (hi << 3);
        const int r1 = r0 + 16;
        float z00 = acc00[v] + bv0;
        float z01 = acc01[v] + bv1;
        float z10 = acc10[v] + bv0;
        float z11 = acc11[v] + bv1;
        if (act == 1) {
            z00 = softplus_f(z00); z01 = softplus_f(z01);
            z10 = softplus_f(z10); z11 = softplus_f(z11);
        }
        C[(size_t)r0 * ldc + c0] = z00;
        C[(size_t)r0 * ldc + c1] = z01;
        C[(size_t)r1 * ldc + c0] = z10;
        C[(size_t)r1 * ldc + c1] = z11;
    }
}

// ---------------------------------------------------------------------------
// Conv1d over the channel-mixing axis (NCH, kernel 3, pad 1) + bias + SiLU.
// out[b, co, h] = silu( sum_{ci,k} x[b, ci, h+k-1] * w[co, ci, k] + bias[co] )
// One thread per output element. b=8, c=64, h=1024.
// ---------------------------------------------------------------------------
__global__ __launch_bounds__(256)
void conv_silu_f32(const float* __restrict__ x,
                   const float* __restrict__ w,
                   const float* __restrict__ bias,
                   float* __restrict__ out) {
    const int idx = blockIdx.x * blockDim.x + threadIdx.x; // 0 .. 524287
    const int h  = idx & 1023;
    const int t  = idx >> 10;
    const int co = t & 63;
    const int b  = t >> 6;

    float acc = bias[co];
    const float* __restrict__ wb = w + (size_t)co * 64 * 3;
    const float* __restrict__ xb = x + ((size_t)b * 64) * 1024 + h;
#pragma unroll 4
    for (int ci = 0; ci < 64; ++ci) {
        const float* xp = xb + (size_t)ci * 1024;
        const float xm = (h > 0)    ? xp[-1] : 0.0f;
        const float xc = xp[0];
        const float xq = (h < 1023) ? xp[1]  : 0.0f;
        acc = fmaf(wb[ci * 3 + 0], xm, acc);
        acc = fmaf(wb[ci * 3 + 1], xc, acc);
        acc = fmaf(wb[ci * 3 + 2], xq, acc);
    }
    const float sig = 1.0f / (1.0f + expf(-acc));
    out[idx] = acc * sig;
}

// ---------------------------------------------------------------------------
// Fused selective scan + gating + skip.
// One wave per (b, e) pair: 32 lanes x 4 states = 128-wide hidden state held
// entirely in VGPRs (the reference would materialize 268 MB of scan state);
// sequential recurrence over L=64; per-step h.C contraction via 5-step
// shfl_xor butterfly (wave32).
//   h[l] = exp(delta*A) * h[l-1] + delta * B[l] * x[l]
//   y[l] = <h[l], C[l]> + D*x[l]
//   out1 = y * silu(x2) + skip
// ---------------------------------------------------------------------------
__global__ __launch_bounds__(256)
void scan_fused_f32(const float* __restrict__ x1c,    // post conv+silu (b,l,e)
                    const float* __restrict__ x2,     // pre-conv x1 (gate)
                    const float* __restrict__ xskip,  // original x
                    const float* __restrict__ delta,  // softplus'd (b,l,e)
                    const float* __restrict__ dbc,    // (b,l,320): [64:192]=B, [192:320]=C
                    const float* __restrict__ A_log,  // (e,128)
                    const float* __restrict__ Dp,     // (e)
                    float* __restrict__ out1) {
    const int wid  = (blockIdx.x * blockDim.x + threadIdx.x) >> 5;
    const int lane = threadIdx.x & 31;
    const int b = wid >> 10;       // 0..7
    const int e = wid & 1023;      // 0..1023
    const int nb = lane << 2;      // 4 states per lane

    float Aa[4], h[4];
#pragma unroll
    for (int j = 0; j < 4; ++j) {
        Aa[j] = -expf(A_log[(size_t)e * 128 + nb + j]);
        h[j] = 0.0f;
    }
    const float Dv = Dp[e];

    for (int l = 0; l < 64; ++l) {
        const size_t r = (size_t)b * 64 + l;
        const float dlt = delta[r * 1024 + e];
        const float xv  = x1c[r * 1024 + e];
        const float* __restrict__ dbcr = dbc + r * 320;
        float s = 0.0f;
#pragma unroll
        for (int j = 0; j < 4; ++j) {
            const float Bv = dbcr[64 + nb + j];
            const float Cv = dbcr[192 + nb + j];
            h[j] = fmaf(expf(dlt * Aa[j]), h[j], dlt * Bv * xv);
            s = fmaf(h[j], Cv, s);
        }
#pragma unroll
        for (int m = 16; m > 0; m >>= 1) s += __shfl_xor(s, m, 32);

        if (lane == 0) {
            const float y   = fmaf(Dv, xv, s);
            const float g   = x2[r * 1024 + e];
            const float sig = 1.0f / (1.0f + expf(-g));
            out1[r * 1024 + e] = fmaf(y, g * sig, xskip[r * 1024 + e]);
        }
    }
}

// ---------------------------------------------------------------------------
// Launch pipeline. Shapes: dim=1024, dt_rank=64, d_state=128, c(=L)=64, b=8.
// Row count M = b*L = 512 for every GEMM. All 32x32 wave tiles divide exactly.
// ---------------------------------------------------------------------------
extern "C" void kernel_launch(void* const* d_in, const int* in_sizes, int n_in,
                              void* d_out, int out_size, void* d_ws, size_t ws_size,
                              hipStream_t stream) {
    const float* x         = (const float*)d_in[0];   // (8,64,1024)
    const float* proj_w    = (const float*)d_in[1];   // (1024,1024)
    const float* proj_b    = (const float*)d_in[2];   // (1024)
    const float* conv_w    = (const float*)d_in[3];   // (64,64,3)
    const float* conv_b    = (const float*)d_in[4];   // (64)
    const float* deltaBC_w = (const float*)d_in[5];   // (320,1024)
    const float* dt_proj_w = (const float*)d_in[6];   // (1024,64)
    const float* dt_proj_b = (const float*)d_in[7];   // (1024)
    const float* A_log     = (const float*)d_in[8];   // (1024,128)
    const float* Dp        = (const float*)d_in[9];   // (1024)
    float* out = (float*)d_out;

    float* ws = (float*)d_ws;
    float* x1    = ws;                 // 512*1024
    float* x1c   = x1    + 524288;     // 512*1024
    float* dbc   = x1c   + 524288;     // 512*320
    float* delta = dbc   + 163840;     // 512*1024
    float* out1  = delta + 524288;     // 512*1024

    // 1) x1 = x @ proj_w^T + proj_b   (512x1024x1024): (512/32)*(1024/32)=512 tiles -> 64 blocks
    wmma_gemm_f32<<<64, 256, 0, stream>>>(x, 1024, proj_w, 1024, proj_b,
                                          x1, 1024, 1024, 32, 0);
    // 2) x1c = silu(conv(x1) + conv_b)
    conv_silu_f32<<<2048, 256, 0, stream>>>(x1, conv_w, conv_b, x1c);
    // 3) dbc = x1c @ deltaBC_w^T      (512x1024x320): 16*10=160 tiles -> 20 blocks
    wmma_gemm_f32<<<20, 256, 0, stream>>>(x1c, 1024, deltaBC_w, 1024, nullptr,
                                          dbc, 320, 1024, 10, 0);
    // 4) delta = softplus(dbc[:, :64] @ dt_proj_w^T + dt_proj_b)  (512x64x1024)
    wmma_gemm_f32<<<64, 256, 0, stream>>>(dbc, 320, dt_proj_w, 64, dt_proj_b,
                                          delta, 1024, 64, 32, 1);
    // 5) fused selective scan + gate + skip   (8192 waves, one per (b,e))
    scan_fused_f32<<<1024, 256, 0, stream>>>(x1c, x1, x, delta, dbc, A_log, Dp, out1);
    // 6) out = out1 @ proj_w^T + proj_b
    wmma_gemm_f32<<<64, 256, 0, stream>>>(out1, 1024, proj_w, 1024, proj_b,
                                          out, 1024, 1024, 32, 0);
}